// AR_67336497267332
// MI455X (gfx1250) — compile-verified
//
#include <hip/hip_runtime.h>
#include <hip/hip_bf16.h>

#define ORDER   24
#define T_OUT   168
#define T_PAD   176      // 11 tiles of 16
#define NT_TILE 11
#define SRC_D   512
#define BATCH   256
#define SEQ     336
#define WIN0    (SEQ - ORDER)   // 312
#define COEF_FLOATS (T_PAD * ORDER + T_PAD)   // coef[176][24] + cvec[176] = 4400

typedef float v2f __attribute__((ext_vector_type(2)));
typedef float v8f __attribute__((ext_vector_type(8)));

// ---------------------------------------------------------------------------
// Kernel 1: build the AR coefficient matrix from W.
//   coef[t][k] : contribution of initial-window element k to output t
//   cvec[t]    : contribution of the bias to output t
// One wave; lane j (j<24) owns column j of the companion state, lane 24 owns
// the bias column. Columns are fully independent -> no synchronization.
// Rows 168..175 are zero padding so kernel 2 can run a full 11th tile.
// ---------------------------------------------------------------------------
__global__ void ar_coef_kernel(const float* __restrict__ W,
                               float* __restrict__ coef,
                               float* __restrict__ cvec) {
    const int j = threadIdx.x;            // 0..31, active 0..24
    float Wreg[ORDER];
#pragma unroll
    for (int k = 0; k < ORDER; ++k) Wreg[k] = W[k];

    float col[ORDER];
#pragma unroll
    for (int k = 0; k < ORDER; ++k)
        col[k] = (j < ORDER && k == j) ? 1.0f : 0.0f;

    for (int t = 0; t < T_OUT; ++t) {
        float a = 0.0f;
#pragma unroll
        for (int k = 0; k < ORDER; ++k) a = fmaf(Wreg[k], col[k], a);
        if (j == ORDER) a += 1.0f;                 // "+ b" injected each step

        if (j < ORDER)       coef[t * ORDER + j] = a;
        else if (j == ORDER) cvec[t] = a;

#pragma unroll
        for (int k = 0; k < ORDER - 1; ++k) col[k] = col[k + 1];
        col[ORDER - 1] = a;
    }
    for (int t = T_OUT; t < T_PAD; ++t) {          // zero padding rows
        if (j < ORDER)       coef[t * ORDER + j] = 0.0f;
        else if (j == ORDER) cvec[t] = 0.0f;
    }
}

// ---------------------------------------------------------------------------
// Kernel 2: Out[b, t, d] = sum_k coef[t,k] * x[b, WIN0+k, d] + cvec[t]*bias
//
// One wave owns a (batch, 16-wide d-slice); it loads the 24x16 x-window into
// registers ONCE (6 v2f B operands), stages coef/cvec in LDS (shared by the
// 8 waves of the block), then loops over 11 t-tiles: 6 chained
// V_WMMA_F32_16X16X4_F32 per tile, fp32 end-to-end.
//
// 32-bit WMMA operand layouts (ISA 7.12.2):
//   A (16x4, MxK): lane l -> M = l&15 ; VGPR0 holds K = 2*(l>>4), VGPR1 K+1
//   B (4x16, KxN): lane l -> N = l&15 ; VGPR0 holds K = 2*(l>>4), VGPR1 K+1
//   C/D (16x16) : VGPR r, lane l -> M = r + 8*(l>>4), N = l&15
//
// Store masking: only t-tile 10 is partial (t=160..175); its valid rows all
// live in lanes 0-15 (hi==0), so the predicate is one uniform half-wave test
// around 8 immediate-offset b32 stores.
// ---------------------------------------------------------------------------
__global__ void __launch_bounds__(256)
ar_wmma_kernel(const float* __restrict__ x,
               const float* __restrict__ coefg,
               const float* __restrict__ bptr,
               float* __restrict__ out) {
    __shared__ float lds[COEF_FLOATS];       // coef[176][24] then cvec[176]

    // cooperative stage of coef+cvec (contiguous in workspace) into LDS
    for (int i = threadIdx.x; i < COEF_FLOATS; i += 256)
        lds[i] = coefg[i];
    __syncthreads();

    const float* scoef = lds;
    const float* scvec = lds + T_PAD * ORDER;

    const int lane = threadIdx.x & 31;
    const int hi   = lane >> 4;                 // 0: lanes 0-15, 1: lanes 16-31
    const int lo   = lane & 15;

    const int tile   = blockIdx.x * 8 + (threadIdx.x >> 5);
    const int d_tile = tile & 31;               // 32 d-tiles
    const int b      = tile >> 5;               // 256 batches

    const float bias = bptr[0];
    const int d = d_tile * 16 + lo;

    // ---- load the 24x16 x-window once: B operands for all 6 K-chunks ----
    const float* xw = x + (long)b * SEQ * SRC_D + (long)WIN0 * SRC_D + d;
    v2f bb[ORDER / 4];
#pragma unroll
    for (int kc = 0; kc < ORDER / 4; ++kc) {
        const int kB = 4 * kc + 2 * hi;
        bb[kc][0] = xw[(kB)     * SRC_D];
        bb[kc][1] = xw[(kB + 1) * SRC_D];
    }

    float* obase = out + (long)b * T_OUT * SRC_D + d;

    for (int t_tile = 0; t_tile < NT_TILE; ++t_tile) {
        const int t0 = t_tile * 16;

        // C init with bias contribution: C[r] = cvec[t0 + r + 8*hi] * bias
        v8f acc;
#pragma unroll
        for (int r = 0; r < 8; ++r)
            acc[r] = scvec[t0 + r + 8 * hi] * bias;

        const int m_row = t0 + lo;              // A-operand row (< 176, padded)
#pragma unroll
        for (int kc = 0; kc < ORDER / 4; ++kc) {
            const int kA = 4 * kc + 2 * hi;
            v2f a;
            a[0] = scoef[m_row * ORDER + kA];
            a[1] = scoef[m_row * ORDER + kA + 1];
            acc = __builtin_amdgcn_wmma_f32_16x16x4_f32(
                      false, a, false, bb[kc], (short)0, acc, false, false);
        }

        // rows t0 + r + 8*hi ; partial only in the last tile, where valid
        // rows are exactly the hi==0 half-wave.
        float* p = obase + (long)(t0 + 8 * hi) * SRC_D;
        if (t_tile < NT_TILE - 1 || hi == 0) {
#pragma unroll
            for (int r = 0; r < 8; ++r)
                p[r * SRC_D] = acc[r];
        }
    }
}

extern "C" void kernel_launch(void* const* d_in, const int* in_sizes, int n_in,
                              void* d_out, int out_size, void* d_ws, size_t ws_size,
                              hipStream_t stream) {
    const float* x  = (const float*)d_in[0];  // [256, 336, 512]
    const float* W  = (const float*)d_in[1];  // [24, 1]
    const float* bp = (const float*)d_in[2];  // [1]
    float* out = (float*)d_out;               // [256, 168, 512]

    float* coef = (float*)d_ws;               // [176][24] then cvec[176]
    float* cvec = coef + T_PAD * ORDER;

    ar_coef_kernel<<<1, 32, 0, stream>>>(W, coef, cvec);

    // 256 batches * 32 d-tiles = 8192 wave-tiles, 8 waves/block
    const int tiles = BATCH * 32;
    ar_wmma_kernel<<<tiles / 8, 256, 0, stream>>>(x, coef, bp, out);
}